// Clustering_80900003987951
// MI455X (gfx1250) — compile-verified
//
#include <hip/hip_runtime.h>

#define B_ 128
#define K_ 64
#define S_ 128
#define F_ 64

#define NBT (B_ / 16)   // 8 b-tiles
#define SC_ 16          // s-chunks (one block each)
#define SS_ (S_ / SC_)  // 8 s-slices per chunk
// per wave: SS_/4 = 2 slices

typedef __attribute__((ext_vector_type(2))) float v2f;
typedef __attribute__((ext_vector_type(4))) float v4f;
typedef __attribute__((ext_vector_type(8))) float v8f;

// ---------------------------------------------------------------------------
// Kernel 1: squared norms  xs2[b,s] = sum_f x^2,  cs2[k,s] = sum_f c^2
// ---------------------------------------------------------------------------
__global__ __launch_bounds__(256)
void norms_kernel(const float* __restrict__ x, const float* __restrict__ c,
                  float* __restrict__ xs2, float* __restrict__ cs2) {
  const int NX = B_ * S_;
  const int NC = K_ * S_;
  int t = blockIdx.x * blockDim.x + threadIdx.x;
  if (t < NX) {
    const v4f* p = (const v4f*)(x + (size_t)t * F_);
    float s = 0.f;
#pragma unroll
    for (int i = 0; i < F_ / 4; ++i) {
      v4f v = p[i];
      s += v.x * v.x + v.y * v.y + v.z * v.z + v.w * v.w;
    }
    xs2[t] = s;
  } else if (t < NX + NC) {
    int j = t - NX;
    const v4f* p = (const v4f*)(c + (size_t)j * F_);
    float s = 0.f;
#pragma unroll
    for (int i = 0; i < F_ / 4; ++i) {
      v4f v = p[i];
      s += v.x * v.x + v.y * v.y + v.z * v.z + v.w * v.w;
    }
    cs2[j] = s;
  }
}

// ---------------------------------------------------------------------------
// Kernel 2: partial dist per (b-tile, s-chunk) via f32 WMMA.
//   dist2[b,k,s] = xs2[b,s] + cs2[k,s] - 2 * dot_f(x[b,s,:], c[k,s,:])
//   partial[bt,sc][b,k] = sum_{s in chunk} sqrt(dist2)
// Grid: (SC_ , NBT).  Block: 512 threads = 16 wave32
//   wave -> (ktile = w&3, s-subchunk = w>>2, 2 slices each)
// ---------------------------------------------------------------------------
__global__ __launch_bounds__(512)
void dist_kernel(const float* __restrict__ x, const float* __restrict__ cent,
                 const float* __restrict__ xs2, const float* __restrict__ cs2,
                 float* __restrict__ partial) {
  __shared__ float distP[4][16][K_];   // 16 KB: per s-subchunk partial tiles

  const int tid   = threadIdx.x;
  const int wave  = tid >> 5;
  const int lane  = tid & 31;
  const int ktile = wave & 3;
  const int ssub  = wave >> 2;
  const int sc = blockIdx.x;           // s-chunk
  const int bt = blockIdx.y;           // b-tile
  const int b0 = bt * 16;
  const int k0 = ktile * 16;

  const int lo  = lane & 15;           // M (A rows) / N (B cols)
  const int hi2 = (lane >> 4) * 2;     // K-pair offset within 4-chunk
  const int hi8 = (lane >> 4) * 8;     // C/D row offset

  const int bA = b0 + lo;              // A-matrix row (b index)
  const int kB = k0 + lo;              // B-matrix col (k index)

  const int sBeg = sc * SS_ + ssub * 2;

  v8f dist = {0.f, 0.f, 0.f, 0.f, 0.f, 0.f, 0.f, 0.f};

#pragma unroll
  for (int si = 0; si < 2; ++si) {
    const int s = sBeg + si;
    const float* xp = x    + ((size_t)bA * S_ + s) * F_ + hi2;
    const float* cp = cent + ((size_t)kB * S_ + s) * F_ + hi2;

    v8f acc = {0.f, 0.f, 0.f, 0.f, 0.f, 0.f, 0.f, 0.f};
#pragma unroll
    for (int f0 = 0; f0 < F_; f0 += 4) {
      v2f a = *(const v2f*)(xp + f0);
      v2f b = *(const v2f*)(cp + f0);
      // D = A(16x4,f32) * B(4x16,f32) + C  -> v_wmma_f32_16x16x4_f32
      acc = __builtin_amdgcn_wmma_f32_16x16x4_f32(
          /*neg_a=*/false, a, /*neg_b=*/false, b,
          /*c_mod=*/(short)0, acc, /*reuse_a=*/false, /*reuse_b=*/false);
    }

    const float c2 = cs2[kB * S_ + s];
#pragma unroll
    for (int r = 0; r < 8; ++r) {
      float x2 = xs2[(b0 + hi8 + r) * S_ + s];
      float d2 = x2 + c2 - 2.0f * acc[r];
      d2 = d2 > 0.f ? d2 : 0.f;        // guard fp round-off
      dist[r] += sqrtf(d2);
    }
  }

  // C/D layout: lane L, VGPR r -> (M = r + 8*(L>=16), N = L&15)
#pragma unroll
  for (int r = 0; r < 8; ++r)
    distP[ssub][hi8 + r][k0 + lo] = dist[r];
  __syncthreads();

  // Combine 4 s-subchunks, emit one disjoint 16x64 partial tile per block.
  float* pout = partial + ((size_t)bt * SC_ + sc) * (16 * K_);
#pragma unroll
  for (int j = 0; j < 2; ++j) {
    int idx = tid + 512 * j;           // 0..1023
    int br  = idx >> 6;
    int kc  = idx & 63;
    pout[idx] = distP[0][br][kc] + distP[1][br][kc] +
                distP[2][br][kc] + distP[3][br][kc];
  }
}

// ---------------------------------------------------------------------------
// Kernel 3: reduce s-chunk partials, Student-t epilogue, normalize over k.
// Grid: NBT blocks of 512 threads.
// ---------------------------------------------------------------------------
__global__ __launch_bounds__(512)
void finalize_kernel(const float* __restrict__ partial, float* __restrict__ out) {
  __shared__ float qTile[16][K_];
  __shared__ float rowsum[16];

  const int bt  = blockIdx.x;
  const int tid = threadIdx.x;
  const float* pin = partial + (size_t)bt * SC_ * (16 * K_);

#pragma unroll
  for (int j = 0; j < 2; ++j) {
    int idx = tid + 512 * j;
    float d = 0.f;
#pragma unroll
    for (int sc = 0; sc < SC_; ++sc) d += pin[sc * (16 * K_) + idx];
    float q = 1.0f / (1.0f + d * d * 0.5f);  // alpha = 2
    q = (q * q * q) * 0.5f;                  // q^(alpha+1) / 2
    qTile[idx >> 6][idx & 63] = q;
  }
  __syncthreads();

  if (tid < 16) {
    float s = 0.f;
#pragma unroll
    for (int kc = 0; kc < K_; ++kc) s += qTile[tid][kc];
    rowsum[tid] = s;
  }
  __syncthreads();

#pragma unroll
  for (int j = 0; j < 2; ++j) {
    int idx = tid + 512 * j;
    int br  = idx >> 6;
    int kc  = idx & 63;
    out[(size_t)(bt * 16 + br) * K_ + kc] = qTile[br][kc] / rowsum[br];
  }
}

// ---------------------------------------------------------------------------
extern "C" void kernel_launch(void* const* d_in, const int* in_sizes, int n_in,
                              void* d_out, int out_size, void* d_ws, size_t ws_size,
                              hipStream_t stream) {
  const float* x = (const float*)d_in[0];   // [B,S,F,1] f32
  const float* c = (const float*)d_in[1];   // [1,K,S,F] f32
  float* xs2     = (float*)d_ws;            // [B*S]  = 16384 floats
  float* cs2     = xs2 + B_ * S_;           // [K*S]  =  8192 floats
  float* partial = cs2 + K_ * S_;           // [NBT*SC_*16*K_] = 131072 floats
  float* out     = (float*)d_out;           // [B,K]

  int total = B_ * S_ + K_ * S_;
  norms_kernel<<<(total + 255) / 256, 256, 0, stream>>>(x, c, xs2, cs2);
  dist_kernel<<<dim3(SC_, NBT), 512, 0, stream>>>(x, c, xs2, cs2, partial);
  finalize_kernel<<<NBT, 512, 0, stream>>>(partial, out);
}